// Radon_84645215469980
// MI455X (gfx1250) — compile-verified
//
#include <hip/hip_runtime.h>
#include <hip/hip_bf16.h>

// Radon forward projection, MI455X (gfx1250, wave32).
// imgs: (2, 512, 512) f32 ; rays: (512, 4) f32 ; angles: (128,) f32
// out:  (2, 128, 512) f32
//
// Gather-issue + VALU bound (2 MB image is L2-resident; HBM traffic ~2.5 MB).
// d_ws holds a zero-padded 514x514 interleaved float2 image so that:
//  (a) each bilinear corner is one b64 gather serving both batch images,
//  (b) the zero border reproduces the reference's valid-mask semantics with
//      no clamp/mask math (samples step at most 1 px outside the image),
//  (c) the 4 corners share one computed address (+1/+PW/+PW+1 folded into
//      load immediate byte offsets),
//  (d) the (img0,img1) pair in each 64-bit load is consumed directly by
//      packed-f32 math with broadcast weights; two accumulator pairs for ILP,
//  (e) prefetch is hoisted to a 16-step outer loop (one uniform branch per
//      chunk), inner loop unrolled 4x (16 loads in flight).

typedef __attribute__((ext_vector_type(2))) float v2f;

constexpr int RES     = 512;
constexpr int NSTEPS  = 512;
constexpr int NANGLES = 128;
constexpr int PW      = RES + 2;   // padded width (1-px zero border each side)
constexpr int CHUNK   = 16;        // prefetch period (steps)

// Build zero-padded interleaved image: pix[(y+1)*PW + (x+1)] = {img0[y][x], img1[y][x]}
__global__ __launch_bounds__(256) void radon_pad_kernel(
    const float* __restrict__ imgs, v2f* __restrict__ pix) {
  const int i = blockIdx.x * blockDim.x + threadIdx.x;
  if (i >= PW * PW) return;
  const int y = i / PW;
  const int x = i - y * PW;
  v2f v = {0.0f, 0.0f};
  if (x >= 1 && x <= RES && y >= 1 && y <= RES) {
    const int src = (y - 1) * RES + (x - 1);
    v.x = imgs[src];
    v.y = imgs[src + RES * RES];
  }
  pix[i] = v;
}

__global__ __launch_bounds__(256) void radon_fwd_padded_kernel(
    const v2f*    __restrict__ pix,
    const float4* __restrict__ rays4,
    const float*  __restrict__ angles,
    float*        __restrict__ out)
{
  const int tid = blockIdx.x * blockDim.x + threadIdx.x;
  const int a   = tid >> 9;          // angle index 0..127
  const int r   = tid & (RES - 1);   // ray index   0..511 (lane-adjacent rays)

  float sn, cs;
  sincosf(angles[a], &sn, &cs);

  const float4 ray = rays4[r];
  const float sx = ray.x, sy = ray.y, ex = ray.z, ey = ray.w;

  const float rsx = cs * sx - sn * sy;
  const float rsy = sn * sx + cs * sy;
  const float rex = cs * ex - sn * ey;
  const float rey = sn * ex + cs * ey;

  const float half  = (float)(RES / 2) - 0.5f;   // 255.5
  const v2f   base  = {rsx + half, rsy + half};  // (bx, by)
  const v2f   delta = {rex - rsx, rey - rsy};    // (dx, dy)
  const float inv_n = 1.0f / (float)NSTEPS;

  v2f accA = {0.0f, 0.0f};   // corners 00 + 01   (batch0, batch1)
  v2f accB = {0.0f, 0.0f};   // corners 10 + 11   (batch0, batch1)

  for (int c = 0; c < NSTEPS; c += CHUNK) {
    // Warm WGP$ ~16 steps ahead, once per chunk (gfx1250 global_prefetch_b8).
    if (c + CHUNK < NSTEPS) {
      const float tp  = ((float)(c + CHUNK) + 0.5f) * inv_n;
      const float pfx = fmaf(tp, delta.x, base.x);
      const float pfy = fmaf(tp, delta.y, base.y);
      const int   pidx = (__float2int_rd(pfy) + 1) * PW + (__float2int_rd(pfx) + 1);
      __builtin_prefetch((const void*)&pix[pidx], 0, 3);
    }

#pragma unroll 4
    for (int j = 0; j < CHUNK; ++j) {
      const int   st = c + j;
      const float t  = ((float)st + 0.5f) * inv_n;

      // (fx, fy) in one packed FMA
      const v2f f = __builtin_elementwise_fma((v2f){t, t}, delta, base);
      const float fx = f.x, fy = f.y;

      const int   x0 = __float2int_rd(fx);          // in [-1, 511]
      const int   y0 = __float2int_rd(fy);          // in [-1, 511]
      const v2f   w  = {__builtin_amdgcn_fractf(fx),
                        __builtin_amdgcn_fractf(fy)};   // (wx, wy)
      const v2f   ow = (v2f){1.0f, 1.0f} - w;           // packed (1-wx, 1-wy)

      // padded base index; corners are idx, idx+1, idx+PW, idx+PW+1
      const int idx = (y0 + 1) * PW + (x0 + 1);     // >= 0 always

      // (w00, w10) and (w01, w11) as two packed muls
      const v2f wa  = {ow.x, w.x};                  // (owx, wx)
      const v2f wr0 = wa * (v2f){ow.y, ow.y};       // (w00, w10)
      const v2f wr1 = wa * (v2f){w.y,  w.y};        // (w01, w11)

      const v2f p00 = pix[idx];
      const v2f p10 = pix[idx + 1];
      const v2f p01 = pix[idx + PW];
      const v2f p11 = pix[idx + PW + 1];

      // two independent packed accumulate chains (both batches per op)
      accA = __builtin_elementwise_fma(p00, (v2f){wr0.x, wr0.x}, accA);
      accB = __builtin_elementwise_fma(p10, (v2f){wr0.y, wr0.y}, accB);
      accA = __builtin_elementwise_fma(p01, (v2f){wr1.x, wr1.x}, accA);
      accB = __builtin_elementwise_fma(p11, (v2f){wr1.y, wr1.y}, accB);
    }
  }

  const v2f   acc   = accA + accB;
  const float scale = (ey - sy) * inv_n;              // length / n, per ray
  out[a * RES + r]                 = acc.x * scale;   // batch 0
  out[NANGLES * RES + a * RES + r] = acc.y * scale;   // batch 1
}

// Fallback (no workspace): clamp+mask path reading the raw images.
__global__ __launch_bounds__(256) void radon_fwd_raw_kernel(
    const float*  __restrict__ im0,
    const float4* __restrict__ rays4,
    const float*  __restrict__ angles,
    float*        __restrict__ out)
{
  const int tid = blockIdx.x * blockDim.x + threadIdx.x;
  const int a   = tid >> 9;
  const int r   = tid & (RES - 1);

  float sn, cs;
  sincosf(angles[a], &sn, &cs);

  const float4 ray = rays4[r];
  const float sx = ray.x, sy = ray.y, ex = ray.z, ey = ray.w;

  const float rsx = cs * sx - sn * sy;
  const float rsy = sn * sx + cs * sy;
  const float rex = cs * ex - sn * ey;
  const float rey = sn * ex + cs * ey;

  const float half  = (float)(RES / 2) - 0.5f;
  const float bx    = rsx + half;
  const float by    = rsy + half;
  const float dx    = rex - rsx;
  const float dy    = rey - rsy;
  const float inv_n = 1.0f / (float)NSTEPS;

  const float* __restrict__ im1 = im0 + RES * RES;

  float acc0 = 0.0f, acc1 = 0.0f;

#pragma unroll 2
  for (int st = 0; st < NSTEPS; ++st) {
    const float t  = ((float)st + 0.5f) * inv_n;
    const float fx = fmaf(t, dx, bx);
    const float fy = fmaf(t, dy, by);
    const int   x0 = __float2int_rd(fx);
    const int   y0 = __float2int_rd(fy);
    const float wx = __builtin_amdgcn_fractf(fx);
    const float wy = __builtin_amdgcn_fractf(fy);
    const int   x1 = x0 + 1, y1 = y0 + 1;

    const float vx0 = ((unsigned)x0 < (unsigned)RES) ? 1.0f : 0.0f;
    const float vx1 = ((unsigned)x1 < (unsigned)RES) ? 1.0f : 0.0f;
    const float vy0 = ((unsigned)y0 < (unsigned)RES) ? 1.0f : 0.0f;
    const float vy1 = ((unsigned)y1 < (unsigned)RES) ? 1.0f : 0.0f;

    const int xc0 = min(max(x0, 0), RES - 1);
    const int xc1 = min(max(x1, 0), RES - 1);
    const int yc0 = min(max(y0, 0), RES - 1);
    const int yc1 = min(max(y1, 0), RES - 1);

    const float w00 = (1.0f - wx) * (1.0f - wy) * (vx0 * vy0);
    const float w10 = wx * (1.0f - wy) * (vx1 * vy0);
    const float w01 = (1.0f - wx) * wy * (vx0 * vy1);
    const float w11 = wx * wy * (vx1 * vy1);

    const int i00 = yc0 * RES + xc0;
    const int i10 = yc0 * RES + xc1;
    const int i01 = yc1 * RES + xc0;
    const int i11 = yc1 * RES + xc1;

    acc0 = fmaf(im0[i00], w00, acc0);
    acc0 = fmaf(im0[i10], w10, acc0);
    acc0 = fmaf(im0[i01], w01, acc0);
    acc0 = fmaf(im0[i11], w11, acc0);
    acc1 = fmaf(im1[i00], w00, acc1);
    acc1 = fmaf(im1[i10], w10, acc1);
    acc1 = fmaf(im1[i01], w01, acc1);
    acc1 = fmaf(im1[i11], w11, acc1);
  }

  const float scale = (ey - sy) * inv_n;
  out[a * RES + r]                 = acc0 * scale;
  out[NANGLES * RES + a * RES + r] = acc1 * scale;
}

extern "C" void kernel_launch(void* const* d_in, const int* in_sizes, int n_in,
                              void* d_out, int out_size, void* d_ws, size_t ws_size,
                              hipStream_t stream) {
  (void)in_sizes; (void)n_in; (void)out_size;

  const float*  imgs   = (const float*)d_in[0];   // (2,512,512) f32
  const float4* rays4  = (const float4*)d_in[1];  // (512,4)     f32
  const float*  angles = (const float*)d_in[2];   // (128,)      f32
  float*        out    = (float*)d_out;           // (2,128,512) f32

  const size_t need  = (size_t)PW * PW * sizeof(v2f);  // ~2.02 MB
  const int    total = NANGLES * RES;                  // 65536 threads

  if (d_ws != nullptr && ws_size >= need) {
    radon_pad_kernel<<<(PW * PW + 255) / 256, 256, 0, stream>>>(
        imgs, (v2f*)d_ws);
    radon_fwd_padded_kernel<<<total / 256, 256, 0, stream>>>(
        (const v2f*)d_ws, rays4, angles, out);
  } else {
    radon_fwd_raw_kernel<<<total / 256, 256, 0, stream>>>(
        imgs, rays4, angles, out);
  }
}